// admmSolver_45165876085273
// MI455X (gfx1250) — compile-verified
//
#include <hip/hip_runtime.h>
#include <hip/hip_bf16.h>

// ADMM solver for MI455X (gfx1250, wave32).
//
// Key structure: the ADMM iteration is column-separable (Tpred[:,n] depends
// only on B[:,n]; z/u updates are elementwise), so a block owning a 16-column
// panel needs no cross-block communication. One persistent kernel runs all
// 1000 iterations with z/u/Atb/B panels resident in LDS (64KB/WGP of the
// 320KB) and the Minv row-panel held in VGPRs (64 x v2f per wave, preloaded
// once). Steady-state inner loop = ds_load_b64 + v_wmma_f32_16x16x4_f32.
//
// Launches per call: prep (WMMA, G & Atb) -> init (Minv=I) -> Gauss-Jordan
// inverse -> persistent ADMM. 4 launches total instead of 1003.

typedef __attribute__((ext_vector_type(2))) float v2f;
typedef __attribute__((ext_vector_type(8))) float v8f;

#define NDIM 256
#define MDIM 2048
#define ITERS 1000

// ---------------------------------------------------------------------------
// prep: G = A^T A (rho*I added here too) and Atb = A^T mic, both 256x256,
// K=2048. 512 output tiles of 16x16; one wave per tile; WMMA f32 16x16x4.
// ---------------------------------------------------------------------------
__global__ __launch_bounds__(256) void prep_kernel(
    const float* __restrict__ sA,     // [2048,256]
    const float* __restrict__ mic,    // [2048,256]
    const float* __restrict__ rho_p,
    float* __restrict__ G,            // [256,256]
    float* __restrict__ Atb)          // [256,256]
{
    const int lane = threadIdx.x & 31;
    const int wave = threadIdx.x >> 5;
    const int gw   = blockIdx.x * 8 + wave;          // 0..511
    const bool isG = (gw < 256);
    const int t    = isG ? gw : gw - 256;
    const int tm   = t >> 4;
    const int tn   = t & 15;
    const int row  = tm * 16 + (lane & 15);          // output row i
    const int col  = tn * 16 + (lane & 15);          // output col j
    const int ksel = (lane >> 4) << 1;               // lanes 0-15: K+{0,1}; 16-31: K+{2,3}
    const float* Bsrc = isG ? sA : mic;

    v8f acc = {};
    for (int k0 = 0; k0 < MDIM; k0 += 4) {
        const int ka = k0 + ksel;
        v2f a, b;
        // A-operand is A^T: A^T[i][m] = sA[m*256 + i]
        a.x = sA[(ka + 0) * NDIM + row];
        a.y = sA[(ka + 1) * NDIM + row];
        b.x = Bsrc[(ka + 0) * NDIM + col];
        b.y = Bsrc[(ka + 1) * NDIM + col];
        acc = __builtin_amdgcn_wmma_f32_16x16x4_f32(
            false, a, false, b, (short)0, acc, false, false);
    }

    const float rho = fabsf(rho_p[0]) + 1e-10f;
    const int mbase = tm * 16 + ((lane >> 4) << 3);
#pragma unroll
    for (int r = 0; r < 8; ++r) {
        const int m  = mbase + r;
        const int gi = m * NDIM + col;
        float v = acc[r];
        if (isG) {
            if (m == col) v += rho;
            G[gi] = v;
        } else {
            Atb[gi] = v;
        }
    }
}

// ---------------------------------------------------------------------------
// init: Minv = I
// ---------------------------------------------------------------------------
__global__ __launch_bounds__(256) void init_kernel(float* __restrict__ Minv)
{
    const int i = blockIdx.x * 256 + threadIdx.x;    // 0..65535
    const int r = i >> 8, c = i & 255;
    Minv[i] = (r == c) ? 1.0f : 0.0f;
}

// ---------------------------------------------------------------------------
// Gauss-Jordan inverse of W (SPD + rho*I -> no pivoting), result in Minv.
// Single workgroup of 256 threads; pivot rows staged in LDS.
// ---------------------------------------------------------------------------
__global__ __launch_bounds__(256) void gj_kernel(
    float* __restrict__ W, float* __restrict__ Minv)
{
    __shared__ float pivW[NDIM];
    __shared__ float pivI[NDIM];
    const int tid = threadIdx.x;

    for (int p = 0; p < NDIM; ++p) {
        __threadfence_block();
        __syncthreads();
        const float ip = 1.0f / W[p * NDIM + p];
        const float wv = W[p * NDIM + tid] * ip;
        const float iv = Minv[p * NDIM + tid] * ip;
        pivW[tid] = wv;
        pivI[tid] = iv;
        W[p * NDIM + tid]    = wv;
        Minv[p * NDIM + tid] = iv;
        __syncthreads();
        if (tid != p) {
            const float f = W[tid * NDIM + p];
            for (int c = 0; c < NDIM; ++c) {
                W[tid * NDIM + c]    -= f * pivW[c];
                Minv[tid * NDIM + c] -= f * pivI[c];
            }
        }
    }
}

// ---------------------------------------------------------------------------
// Persistent ADMM kernel: 16 blocks x 512 threads; block b owns columns
// [16b, 16b+16). All 1000 iterations run inside the kernel; z/u/Atb/B panels
// live in LDS; the Minv row-panel for each wave lives in VGPRs.
//
// Per iteration:
//   sync; B = Atb + rho*(z - u)         (LDS->LDS, K-paired layout)
//   sync; Tpred_tile = Minv_panel @ B   (64 x v_wmma_f32_16x16x4_f32 / wave)
//   z,u <- soft-threshold/clip updates  (LDS, wave-private locations)
// ---------------------------------------------------------------------------
__global__ __launch_bounds__(512) void admm_persistent_kernel(
    const float* __restrict__ Minv,
    const float* __restrict__ Atb,
    float* __restrict__ Tpred,
    const float* __restrict__ rho_p,
    const float* __restrict__ lam_p)
{
    __shared__ float AtbS[NDIM * 16];   // [k][n] flat
    __shared__ float Zs[NDIM * 16];     // [k][n] flat
    __shared__ float Us[NDIM * 16];     // [k][n] flat
    __shared__ float BsP[NDIM * 16];    // K-paired: [(k>>1)*32 + 2n + (k&1)]

    const int tid     = threadIdx.x;    // 0..511
    const int lane    = tid & 31;
    const int wave    = tid >> 5;       // 0..15 == row tile tm
    const int colbase = blockIdx.x * 16;

    const float rho = fabsf(rho_p[0]) + 1e-10f;
    const float tau = fabsf(lam_p[0]) / rho;

    // Stage Atb panel into LDS; zero z/u state.
#pragma unroll
    for (int i = 0; i < 8; ++i) {
        const int f = tid + i * 512;            // 0..4095
        const int k = f >> 4, n = f & 15;
        AtbS[f] = Atb[k * NDIM + colbase + n];
        Zs[f] = 0.0f;
        Us[f] = 0.0f;
    }

    // Preload this wave's Minv row-panel (A-operand stream) into VGPRs:
    // 64 k-steps x v2f = 128 VGPRs, loop-invariant for all 1000 iterations.
    const int row  = wave * 16 + (lane & 15);
    const int ksel = (lane >> 4) << 1;          // lanes 0-15: K+{0,1}; 16-31: K+{2,3}
    v2f Areg[64];
#pragma unroll
    for (int kk = 0; kk < 64; ++kk) {
        Areg[kk] = *(const v2f*)(Minv + row * NDIM + kk * 4 + ksel);
    }

    const int nloc  = lane & 15;
    const int mbase = wave * 16 + ((lane >> 4) << 3);

    for (int it = 0; it < ITERS; ++it) {
        __syncthreads();                        // z/u updates visible
        // Build B panel in K-paired layout (ds_load_b64-friendly).
#pragma unroll
        for (int i = 0; i < 8; ++i) {
            const int f = tid + i * 512;
            const int k = f >> 4, n = f & 15;
            const float b = AtbS[f] + rho * (Zs[f] - Us[f]);
            BsP[((k >> 1) << 5) + (n << 1) + (k & 1)] = b;
        }
        __syncthreads();                        // B panel complete

        v8f acc = {};
#pragma unroll
        for (int kk = 0; kk < 64; ++kk) {
            const int ka = kk * 4 + ksel;       // even -> 8B aligned
            const v2f b = *(const v2f*)&BsP[((ka >> 1) << 5) + (nloc << 1)];
            acc = __builtin_amdgcn_wmma_f32_16x16x4_f32(
                false, Areg[kk], false, b, (short)0, acc, false, false);
        }

        // Fused epilogue: soft-threshold + clip + dual update (wave-private).
#pragma unroll
        for (int r = 0; r < 8; ++r) {
            const int m  = mbase + r;
            const int li = m * 16 + nloc;
            const float d  = acc[r];
            const float uo = Us[li];
            const float g  = d + uo;
            float zt = fmaxf(g - tau, 0.0f) + fminf(g + tau, 0.0f);
            zt = fminf(fmaxf(zt, 0.0f), 1.0f);
            Zs[li] = zt;
            Us[li] = uo + d - zt;
            if (it == ITERS - 1) {              // uniform branch
                Tpred[m * NDIM + colbase + nloc] = d;
            }
        }
    }
}

// ---------------------------------------------------------------------------
// Host-side orchestration (graph-capturable: launches only).
// Inputs: 0:T[256,256] 1:s_A[2048,256] 2:s_mic_data[2048,256] 3:rho 4:lam
// ---------------------------------------------------------------------------
extern "C" void kernel_launch(void* const* d_in, const int* in_sizes, int n_in,
                              void* d_out, int out_size, void* d_ws, size_t ws_size,
                              hipStream_t stream) {
    const float* sA    = (const float*)d_in[1];
    const float* mic   = (const float*)d_in[2];
    const float* rho_p = (const float*)d_in[3];
    const float* lam_p = (const float*)d_in[4];
    float* Tp = (float*)d_out;

    float* ws   = (float*)d_ws;
    float* W    = ws + 0 * 65536;   // working copy of G (destroyed by GJ)
    float* Minv = ws + 1 * 65536;
    float* Atb  = ws + 2 * 65536;   // 768 KB of f32 scratch total

    prep_kernel<<<64, 256, 0, stream>>>(sA, mic, rho_p, W, Atb);
    init_kernel<<<256, 256, 0, stream>>>(Minv);
    gj_kernel<<<1, 256, 0, stream>>>(W, Minv);
    admm_persistent_kernel<<<16, 512, 0, stream>>>(Minv, Atb, Tp, rho_p, lam_p);
}